// IterativeFixedPoint_3848290697534
// MI455X (gfx1250) — compile-verified
//
#include <hip/hip_runtime.h>

typedef __attribute__((ext_vector_type(2))) float v2f;
typedef __attribute__((ext_vector_type(8))) float v8f;

#define FEAT 256
#define NT   16            // 256/16 column tiles
#define KC   64            // 256/4  K chunks
#define RPW  16            // rows per wave
#define WPB  4             // waves per workgroup
#define RPB  (RPW * WPB)   // 64 rows per workgroup
#define W_FLOATS (FEAT * FEAT)          // 65536 floats = 256 KB
#define Z_FLOATS (RPW * FEAT)           // 4096 floats = 16 KB per wave
#define LDS_FLOATS (W_FLOATS + WPB * Z_FLOATS)   // 81920 floats = 320 KB
#define ITERS 25

// tanh(y) = 1 - 2/(exp2(2*log2(e)*y) + 1); saturates correctly for |y| large.
__device__ __forceinline__ float fast_tanh(float y) {
    float e = __builtin_amdgcn_exp2f(y * 2.8853900817779268f);
    return 1.0f - 2.0f * __builtin_amdgcn_rcpf(e + 1.0f);
}

__global__ __launch_bounds__(WPB * 32, 1)
void IterativeFixedPoint_kernel(const float* __restrict__ x,
                                const float* __restrict__ W,
                                const float* __restrict__ b,
                                float* __restrict__ out)
{
    extern __shared__ float lds[];
    const int tid  = threadIdx.x;
    const int lane = tid & 31;
    const int wave = tid >> 5;
    const int h    = lane >> 4;   // lane half (0/1)
    const int l16  = lane & 15;

    // ---- Stage W into LDS as the WMMA B operand: B[k][n] = W[n][k].
    // Pair-interleave over k so one ds_load_b64 yields (K, K+1) into 2 VGPRs;
    // XOR-swizzle n by 16*((k>>1)&1) so the two lane-halves use disjoint banks.
    // element (n,k) -> lds[(k>>1)*512 + ((n ^ (((k>>1)&1)<<4)) << 1) + (k&1)]
    #pragma unroll 4
    for (int i = 0; i < W_FLOATS / (WPB * 32); ++i) {
        int f = i * (WPB * 32) + tid;      // consecutive k across threads: coalesced
        int n = f >> 8, k = f & (FEAT - 1);
        int kp = k >> 1;
        lds[kp * 512 + ((n ^ ((kp & 1) << 4)) << 1) + (k & 1)] = W[f];
    }

    float* zbuf = lds + W_FLOATS + wave * Z_FLOATS;    // this wave's 16x256 z tile
    const long rowBase = (long)blockIdx.x * RPB + wave * RPW;
    const float* xw = x + rowBase * FEAT;

    // ---- xb = x + b staged into registers in D-layout (iteration-invariant):
    // xb[nt][v] holds (m = 8*h + v, n = 16*nt + l16)
    v8f xb[NT];
    #pragma unroll
    for (int nt = 0; nt < NT; ++nt) {
        float bn = b[nt * 16 + l16];
        #pragma unroll
        for (int v = 0; v < 8; ++v) {
            int m = 8 * h + v;
            xb[nt][v] = xw[m * FEAT + nt * 16 + l16] + bn;
        }
    }

    // ---- Iteration 1: z = tanh(x + b) (z0 = 0), stored bank-swizzled:
    // (m,k) -> zbuf[m*FEAT + (k ^ (m<<2))]
    #pragma unroll 4
    for (int i = 0; i < Z_FLOATS / 32; ++i) {
        int f = i * 32 + lane;
        int m = f >> 8, k = f & (FEAT - 1);
        zbuf[m * FEAT + (k ^ (m << 2))] = fast_tanh(xw[f] + b[k]);
    }

    __syncthreads();   // W fully staged (z tiles are wave-private)

    // ---- 24 more fixed-point iterations, all data in LDS/registers.
    for (int it = 1; it < ITERS; ++it) {
        v8f acc[NT];
        #pragma unroll
        for (int nt = 0; nt < NT; ++nt) acc[nt] = xb[nt];   // seed with x + b

        #pragma unroll 2
        for (int kc = 0; kc < KC; ++kc) {
            // A operand (16x4 f32): lane(h,l16) holds z[m=l16, K=4kc+2h .. +1]
            v2f A = *(const v2f*)(zbuf + l16 * FEAT + ((4 * kc + 2 * h) ^ (l16 << 2)));
            int kp = 2 * kc + h;
            const float* brow = lds + kp * 512;
            int nsw = (kp & 1) << 4;
            #pragma unroll
            for (int nt = 0; nt < NT; ++nt) {
                // B operand (4x16 f32): VGPR v -> K = 4kc + 2h + v, N = 16*nt + l16
                v2f B = *(const v2f*)(brow + (((nt * 16 + l16) ^ nsw) << 1));
                acc[nt] = __builtin_amdgcn_wmma_f32_16x16x4_f32(
                    false, A, false, B, (short)0, acc[nt], false, false);
            }
        }

        // Epilogue: z_new = tanh(acc); write back swizzled. All LDS reads of the
        // old z happened above; per-wave DS ops are in-order, so no race.
        #pragma unroll
        for (int nt = 0; nt < NT; ++nt) {
            #pragma unroll
            for (int v = 0; v < 8; ++v) {
                int m = 8 * h + v;
                int n = nt * 16 + l16;
                zbuf[m * FEAT + (n ^ (m << 2))] = fast_tanh(acc[nt][v]);
            }
        }
    }

    // ---- Write result (de-swizzle, coalesced stores)
    float* ow = out + rowBase * FEAT;
    #pragma unroll 4
    for (int i = 0; i < Z_FLOATS / 32; ++i) {
        int f = i * 32 + lane;
        int m = f >> 8, k = f & (FEAT - 1);
        ow[f] = zbuf[m * FEAT + (k ^ (m << 2))];
    }
}

extern "C" void kernel_launch(void* const* d_in, const int* in_sizes, int n_in,
                              void* d_out, int out_size, void* d_ws, size_t ws_size,
                              hipStream_t stream) {
    const float* x = (const float*)d_in[0];
    const float* W = (const float*)d_in[1];
    const float* b = (const float*)d_in[2];
    float* out = (float*)d_out;

    const int rows   = in_sizes[0] / FEAT;      // 32768
    const int blocks = rows / RPB;              // 512
    const size_t lds_bytes = LDS_FLOATS * sizeof(float);   // 327680 B = 320 KB

    (void)hipFuncSetAttribute((const void*)IterativeFixedPoint_kernel,
                              hipFuncAttributeMaxDynamicSharedMemorySize,
                              (int)lds_bytes);

    IterativeFixedPoint_kernel<<<blocks, WPB * 32, lds_bytes, stream>>>(x, W, b, out);
}